// CustomRNNCell_90649579749584
// MI455X (gfx1250) — compile-verified
//
#include <hip/hip_runtime.h>

// ---------------------------------------------------------------------------
// Problem dims (fixed by the reference)
// ---------------------------------------------------------------------------
#define B_DIM 512
#define A_DIM 256
#define I_DIM 64
#define S_DIM 512
#define P_DIM 256

typedef __attribute__((ext_vector_type(2))) float v2f;
typedef __attribute__((ext_vector_type(8))) float v8f;

// ---------------------------------------------------------------------------
// f32 WMMA 16x16 tile accumulate:  acc += A[row0:row0+16, :K] * B(:K, col0:col0+16)
//
// Per CDNA5 ISA 7.12.2:
//   A 16x4 f32 frag = 2 VGPRs: lane(l<16) holds {K=0,K=1} of row M=l,
//                              lane(l>=16) holds {K=2,K=3} of row M=l-16.
//   B 4x16 frag mirrored with N = lane&15.
//   C/D 16x16 f32 = 8 VGPRs: element [r + 8*(l>>4), l&15] in vgpr r.
// TRANSB=true reads Bm as row-major [N,K] (i.e. computes A @ Bm^T).
// ---------------------------------------------------------------------------
template <bool TRANSB>
__device__ inline v8f wmma_gemm_accum(v8f acc,
                                      const float* __restrict__ A, int lda,
                                      const float* __restrict__ Bm, int ldb,
                                      int K, int row0, int col0, int lane) {
    const int rc   = lane & 15;        // row of A frag / col of B frag
    const int half = lane >> 4;        // selects K pair {0,1} vs {2,3}
    const float* arow = A + (size_t)(row0 + rc) * lda + 2 * half;
    const float* brow;
    if (TRANSB) {
        brow = Bm + (size_t)(col0 + rc) * ldb + 2 * half;   // contiguous K pair
    } else {
        brow = Bm + (size_t)(2 * half) * ldb + (col0 + rc); // strided K pair
    }
#pragma unroll 4
    for (int k0 = 0; k0 < K; k0 += 4) {
        v2f a = *(const v2f*)(arow + k0);
        v2f b;
        if (TRANSB) {
            b = *(const v2f*)(brow + k0);
        } else {
            const float* bp = brow + (size_t)k0 * ldb;
            b.x = bp[0];
            b.y = bp[ldb];
        }
        acc = __builtin_amdgcn_wmma_f32_16x16x4_f32(
            /*neg_a=*/false, a, /*neg_b=*/false, b,
            /*c_mod=*/(short)0, acc, /*reuse_a=*/false, /*reuse_b=*/false);
    }
    return acc;
}

__device__ inline void wmma_store_tile(float* __restrict__ C, int ldc,
                                       int row0, int col0, int lane, v8f acc) {
    const int col = col0 + (lane & 15);
    const int row = row0 + 8 * (lane >> 4);
#pragma unroll
    for (int r = 0; r < 8; ++r)
        C[(size_t)(row + r) * ldc + col] = acc[r];
}

// Generic GEMM: C[M,N] = A[M,K] @ B  (B row-major [K,N], or [N,K] if TRANSB)
// block = 256 threads = 8 waves; wave tile 16x16; block tile 128(M) x 16(N).
template <bool TRANSB>
__global__ void gemm_wmma_f32(const float* __restrict__ A,
                              const float* __restrict__ Bm,
                              float* __restrict__ C,
                              int M, int N, int K, int lda, int ldb, int ldc) {
    const int lane = threadIdx.x & 31;
    const int wave = threadIdx.x >> 5;
    const int row0 = blockIdx.x * 128 + wave * 16;
    const int col0 = blockIdx.y * 16;
    if (row0 >= M || col0 >= N) return;
    v8f acc = {};
    acc = wmma_gemm_accum<TRANSB>(acc, A, lda, Bm, ldb, K, row0, col0, lane);
    wmma_store_tile(C, ldc, row0, col0, lane, acc);
}

// ---------------------------------------------------------------------------
// Fused new_state:  ns = prev@T1 + dt@T2 + t3 + act@T4 + inp@T3Pm
// One accumulator tile kept live across all four K-phases; bias preloaded.
// ---------------------------------------------------------------------------
__global__ void new_state_kernel(const float* __restrict__ prev,
                                 const float* __restrict__ T1,
                                 const float* __restrict__ dt,
                                 const float* __restrict__ T2,
                                 const float* __restrict__ act,
                                 const float* __restrict__ T4,
                                 const float* __restrict__ inp,
                                 const float* __restrict__ T3Pm,
                                 const float* __restrict__ t3,
                                 float* __restrict__ ns) {
    const int lane = threadIdx.x & 31;
    const int wave = threadIdx.x >> 5;
    const int row0 = blockIdx.x * 128 + wave * 16;
    const int col0 = blockIdx.y * 16;
    const float bias = t3[col0 + (lane & 15)];
    v8f acc;
#pragma unroll
    for (int r = 0; r < 8; ++r) acc[r] = bias;
    acc = wmma_gemm_accum<false>(acc, prev, S_DIM, T1,   S_DIM, S_DIM, row0, col0, lane);
    acc = wmma_gemm_accum<false>(acc, dt,   A_DIM, T2,   S_DIM, A_DIM, row0, col0, lane);
    acc = wmma_gemm_accum<false>(acc, act,  A_DIM, T4,   S_DIM, A_DIM, row0, col0, lane);
    acc = wmma_gemm_accum<false>(acc, inp,  P_DIM, T3Pm, S_DIM, P_DIM, row0, col0, lane);
    wmma_store_tile(ns, S_DIM, row0, col0, lane, acc);
}

// ---------------------------------------------------------------------------
// Piecewise-linear basis prep.
//  X @ w_recover  == first difference along I
//  X @ b_recover  == exclusive prefix sum along I
// One thread per action unit a (A=256 threads).
// ---------------------------------------------------------------------------
__global__ void prep_weights_kernel(const float* __restrict__ wpt,
                                    const float* __restrict__ bpt,
                                    const float* __restrict__ wmt,
                                    const float* __restrict__ bmt,
                                    float* __restrict__ wp, float* __restrict__ bp,
                                    float* __restrict__ wm, float* __restrict__ bm) {
    const int a = blockIdx.x * blockDim.x + threadIdx.x;
    if (a >= A_DIM) return;
    const int base = a * I_DIM;
    float prev_p = 0.f, prev_m = 0.f, cum_p = 0.f, cum_m = 0.f;
    for (int i = 0; i < I_DIM; ++i) {
        float w2 = wpt[base + i]; w2 *= w2;
        wp[base + i] = w2 - prev_p;            // (t^2) @ w_recover
        prev_p = w2;
        float m2 = wmt[base + i]; m2 *= m2;
        wm[base + i] = -(m2 - prev_m);         // (-(t^2)) @ w_recover
        prev_m = m2;
        float b2 = bpt[base + i]; b2 *= b2;
        bp[base + i] = -cum_p;                 // (-(t^2)) @ b_recover (excl cumsum)
        cum_p += b2;
        float c2 = bmt[base + i]; c2 *= c2;
        bm[base + i] = -cum_m;
        cum_m += c2;
    }
}

// ---------------------------------------------------------------------------
// action[b,a] = clip( sum_i relu(f+bp)*wp + relu(-f+bm)*wm , [-max,max] )
// grid.x = a, grid.y tiles b; per-a basis staged in LDS.
// ---------------------------------------------------------------------------
__global__ void action_kernel(const float* __restrict__ freq,
                              const float* __restrict__ wp, const float* __restrict__ bp,
                              const float* __restrict__ wm, const float* __restrict__ bm,
                              const float* __restrict__ max_action,
                              float* __restrict__ action_out) {
    __shared__ float swp[I_DIM], sbp[I_DIM], swm[I_DIM], sbm[I_DIM];
    const int a = blockIdx.x;
    const int t = threadIdx.x;
    if (t < I_DIM) {
        swp[t] = wp[a * I_DIM + t];
        sbp[t] = bp[a * I_DIM + t];
        swm[t] = wm[a * I_DIM + t];
        sbm[t] = bm[a * I_DIM + t];
    }
    __syncthreads();
    const int b = blockIdx.y * blockDim.x + t;
    const float f = freq[(size_t)b * A_DIM + a];
    float acc = 0.f;
#pragma unroll 8
    for (int i = 0; i < I_DIM; ++i)
        acc += fmaxf(f + sbp[i], 0.f) * swp[i] + fmaxf(-f + sbm[i], 0.f) * swm[i];
    const float ma = max_action[0];
    const float act = ma - fmaxf(ma - acc, 0.f) + fmaxf(-ma - acc, 0.f);
    action_out[(size_t)b * A_DIM + a] = act;
}

// sin/cos of delta (feeds the factored delta_term GEMMs)
__global__ void sincos_kernel(const float* __restrict__ delta,
                              float* __restrict__ s, float* __restrict__ c, int n) {
    const int i = blockIdx.x * blockDim.x + threadIdx.x;
    if (i >= n) return;
    const float d = delta[i];
    s[i] = __sinf(d);
    c[i] = __cosf(d);
}

// delta_term = s * (c@F^T) - c * (s@F^T)
__global__ void delta_term_kernel(const float* __restrict__ s,
                                  const float* __restrict__ c,
                                  const float* __restrict__ Cs,
                                  const float* __restrict__ Ss,
                                  float* __restrict__ dt, int n) {
    const int i = blockIdx.x * blockDim.x + threadIdx.x;
    if (i >= n) return;
    dt[i] = s[i] * Cs[i] - c[i] * Ss[i];
}

// loss0[b] = sum_s ns[b,s]^2 * w[s]; one wave32 per batch row.
__global__ void loss_kernel(const float* __restrict__ ns,
                            const float* __restrict__ w,
                            float* __restrict__ loss) {
    const int lane = threadIdx.x & 31;
    const int wave = threadIdx.x >> 5;
    const int b = blockIdx.x * 8 + wave;
    float acc = 0.f;
    for (int sidx = lane; sidx < S_DIM; sidx += 32) {
        const float v = ns[(size_t)b * S_DIM + sidx];
        acc += v * v * w[sidx];
    }
#pragma unroll
    for (int off = 16; off > 0; off >>= 1)
        acc += __shfl_xor(acc, off, 32);
    if (lane == 0) loss[b] = acc;
}

// ---------------------------------------------------------------------------
// Launch
// ---------------------------------------------------------------------------
extern "C" void kernel_launch(void* const* d_in, const int* in_sizes, int n_in,
                              void* d_out, int out_size, void* d_ws, size_t ws_size,
                              hipStream_t stream) {
    const float* inputs       = (const float*)d_in[0];   // [B,P]
    const float* prev_output  = (const float*)d_in[1];   // [B,S]
    const float* w_plus_t0    = (const float*)d_in[2];   // [A,I]
    const float* b_plus_t0    = (const float*)d_in[3];
    const float* w_minus_t0   = (const float*)d_in[4];
    const float* b_minus_t0   = (const float*)d_in[5];
    const float* st1          = (const float*)d_in[6];   // [S,S]
    const float* stF          = (const float*)d_in[7];   // [A,A]
    const float* st2          = (const float*)d_in[8];   // [A,S]
    const float* st3          = (const float*)d_in[9];   // [S]
    const float* st4          = (const float*)d_in[10];  // [A,S]
    const float* st3_Pm       = (const float*)d_in[11];  // [P,S]
    const float* sel_add_w    = (const float*)d_in[12];  // [S,1]
    const float* sel_w        = (const float*)d_in[13];  // [S,A]
    const float* sel_delta    = (const float*)d_in[14];  // [S,A]
    const float* max_action   = (const float*)d_in[15];  // scalar
    // d_in[16..18]: w_recover / b_recover / ones_frequency — folded analytically.

    // Workspace layout (floats)
    float* ws   = (float*)d_ws;
    float* wp   = ws;                         // [A,I]
    float* bp   = wp + A_DIM * I_DIM;
    float* wm   = bp + A_DIM * I_DIM;
    float* bm   = wm + A_DIM * I_DIM;
    float* freq = bm + A_DIM * I_DIM;         // [B,A]
    float* delt = freq + B_DIM * A_DIM;       // [B,A]
    float* sv   = delt + B_DIM * A_DIM;       // sin(delta)
    float* cv   = sv + B_DIM * A_DIM;         // cos(delta)
    float* Cs   = cv + B_DIM * A_DIM;         // c @ F^T
    float* Ss   = Cs + B_DIM * A_DIM;         // s @ F^T
    float* dt   = Ss + B_DIM * A_DIM;         // delta_term

    // Output layout: loss0[B] | frequency[B,A] | action[B,A] | new_state[B,S]
    float* out_loss = (float*)d_out;
    float* out_freq = out_loss + B_DIM;
    float* out_act  = out_freq + B_DIM * A_DIM;
    float* out_ns   = out_act + B_DIM * A_DIM;

    // 1) basis params (difference / exclusive cumsum forms)
    prep_weights_kernel<<<1, 256, 0, stream>>>(w_plus_t0, b_plus_t0, w_minus_t0,
                                               b_minus_t0, wp, bp, wm, bm);

    // 2) freq = prev @ select_w ; delta = prev @ select_delta   [512x512x256]
    gemm_wmma_f32<false><<<dim3(4, 16), 256, 0, stream>>>(
        prev_output, sel_w, freq, B_DIM, A_DIM, S_DIM, S_DIM, A_DIM, A_DIM);
    gemm_wmma_f32<false><<<dim3(4, 16), 256, 0, stream>>>(
        prev_output, sel_delta, delt, B_DIM, A_DIM, S_DIM, S_DIM, A_DIM, A_DIM);

    // 3) piecewise-linear action (written straight into its d_out slot)
    action_kernel<<<dim3(A_DIM, B_DIM / 256), 256, 0, stream>>>(
        freq, wp, bp, wm, bm, max_action, out_act);

    // 4) sin/cos(delta), then factored delta_term GEMMs: Cs = c@F^T, Ss = s@F^T
    sincos_kernel<<<(B_DIM * A_DIM) / 256, 256, 0, stream>>>(delt, sv, cv, B_DIM * A_DIM);
    gemm_wmma_f32<true><<<dim3(4, 16), 256, 0, stream>>>(
        cv, stF, Cs, B_DIM, A_DIM, A_DIM, A_DIM, A_DIM, A_DIM);
    gemm_wmma_f32<true><<<dim3(4, 16), 256, 0, stream>>>(
        sv, stF, Ss, B_DIM, A_DIM, A_DIM, A_DIM, A_DIM, A_DIM);
    delta_term_kernel<<<(B_DIM * A_DIM) / 256, 256, 0, stream>>>(
        sv, cv, Cs, Ss, dt, B_DIM * A_DIM);

    // 5) fused new_state = prev@T1 + dt@T2 + t3 + act@T4 + inp@T3Pm
    new_state_kernel<<<dim3(4, 32), 256, 0, stream>>>(
        prev_output, st1, dt, st2, out_act, st4, inputs, st3_Pm, st3, out_ns);

    // 6) frequency = new_state @ select_w   [512x512x256]
    gemm_wmma_f32<false><<<dim3(4, 16), 256, 0, stream>>>(
        out_ns, sel_w, out_freq, B_DIM, A_DIM, S_DIM, S_DIM, A_DIM, A_DIM);

    // 7) loss0 = (ns^2) @ select_add_w
    loss_kernel<<<B_DIM / 8, 256, 0, stream>>>(out_ns, sel_add_w, out_loss);
}